// JittedGnnModel_34548716929234
// MI455X (gfx1250) — compile-verified
//
#include <hip/hip_runtime.h>

typedef __attribute__((ext_vector_type(2))) float v2f;
typedef __attribute__((ext_vector_type(8))) float v8f;

#define NODES  50000
#define TSTEPS 8
#define FDIM   64
#define SPAN   (TSTEPS * FDIM)   // 512 floats per node per feature map

// ---------------- degree / normalization ----------------
__global__ void k_init_deg(float* deg, int n) {
    int i = blockIdx.x * blockDim.x + threadIdx.x;
    if (i < n) deg[i] = 1.0f;                       // self-loop contributes 1
}

__global__ void k_accum_deg(const long long* __restrict__ cols, float* deg, int e) {
    int i = blockIdx.x * blockDim.x + threadIdx.x;
    if (i < e) atomicAdd(&deg[(int)cols[i]], 1.0f);
}

__global__ void k_rsqrt(float* deg, int n) {
    int i = blockIdx.x * blockDim.x + threadIdx.x;
    if (i < n) deg[i] = rsqrtf(deg[i]);             // deg >= 1 always
}

// ---------------- H[M,64] = X[M,64] @ W[64,64] via V_WMMA_F32_16X16X4_F32 ----------------
// One wave computes a 16-row x 64-col strip: 4 accumulators, 16 K-steps of K=4.
// W is staged into LDS in K-pair-interleaved layout:
//   sW[k2*128 + n*2 + j] == W[(2*k2 + j)*64 + n]
// so each lane's B fragment {W[k][n], W[k+1][n]} is one aligned ds_load_b64.
__global__ void __launch_bounds__(128)
k_gemm64(const float* __restrict__ X, const float* __restrict__ W,
         float* __restrict__ H, int M) {
    __shared__ float sW[64 * 64];
    for (int idx = threadIdx.x; idx < 32 * 64; idx += 128) {
        int k2 = idx >> 6;          // 0..31
        int n  = idx & 63;          // 0..63
        sW[k2 * 128 + n * 2 + 0] = W[(2 * k2 + 0) * 64 + n];
        sW[k2 * 128 + n * 2 + 1] = W[(2 * k2 + 1) * 64 + n];
    }
    __syncthreads();

    const int lane     = threadIdx.x & 31;
    const int wave     = threadIdx.x >> 5;
    const int row_base = (blockIdx.x * 4 + wave) * 16;   // wave-uniform
    if (row_base + 16 <= M) {
        const int m     = lane & 15;           // A row within tile / B-N col within tile
        const int half  = lane >> 4;           // 0 or 1
        const int khalf = half * 2;            // K sub-offset per half-wave
        const float* xrow = X + (size_t)(row_base + m) * 64;

        v8f c0 = {}, c1 = {}, c2 = {}, c3 = {};
#pragma unroll
        for (int kb = 0; kb < 64; kb += 4) {
            // A fragment: lane holds X[row, kb+khalf .. kb+khalf+1]
            v2f a = *(const v2f*)(xrow + kb + khalf);
            // B fragments: one ds_load_b64 per 16-wide col tile
            const float* wp = &sW[(kb / 2 + half) * 128 + m * 2];
            v2f b0 = *(const v2f*)(wp + 0);
            v2f b1 = *(const v2f*)(wp + 32);
            v2f b2 = *(const v2f*)(wp + 64);
            v2f b3 = *(const v2f*)(wp + 96);
            c0 = __builtin_amdgcn_wmma_f32_16x16x4_f32(false, a, false, b0, (short)0, c0, false, false);
            c1 = __builtin_amdgcn_wmma_f32_16x16x4_f32(false, a, false, b1, (short)0, c1, false, false);
            c2 = __builtin_amdgcn_wmma_f32_16x16x4_f32(false, a, false, b2, (short)0, c2, false, false);
            c3 = __builtin_amdgcn_wmma_f32_16x16x4_f32(false, a, false, b3, (short)0, c3, false, false);
        }

        // D layout: VGPR i -> row i + 8*(lane>>4), col = lane&15 (per 16-wide tile)
        const int ro = half * 8;
#pragma unroll
        for (int i = 0; i < 8; ++i) {
            float* o = H + (size_t)(row_base + ro + i) * 64 + m;
            o[0]  = c0[i];
            o[16] = c1[i];
            o[32] = c2[i];
            o[48] = c3[i];
        }
    }
}

// ---------------- agg[n] = h[n] * dinv[n]^2  (self-loop term; fully initializes agg) ----
__global__ void k_selfloop(const float* __restrict__ h, const float* __restrict__ dinv,
                           float* __restrict__ agg, size_t total4) {
    size_t i = (size_t)blockIdx.x * blockDim.x + threadIdx.x;
    if (i >= total4) return;
    size_t idx  = i * 4;
    int    node = (int)(idx >> 9);               // / SPAN (=512)
    float  s    = dinv[node]; s *= s;
    float4 v = *(const float4*)(h + idx);
    v.x *= s; v.y *= s; v.z *= s; v.w *= s;
    *(float4*)(agg + idx) = v;
}

// ---------------- per-edge scatter: one wave per edge, 512 floats ----------------
__global__ void __launch_bounds__(256)
k_edge(const long long* __restrict__ rows, const long long* __restrict__ cols,
       const float* __restrict__ dinv, const float* __restrict__ h,
       float* __restrict__ agg, int E) {
    int w    = blockIdx.x * 8 + (threadIdx.x >> 5);  // wave-uniform edge id
    int lane = threadIdx.x & 31;
    if (w >= E) return;
    int   r = (int)rows[w];
    int   c = (int)cols[w];
    float s = dinv[r] * dinv[c];
    const float* src = h   + (size_t)r * SPAN;
    float*       dst = agg + (size_t)c * SPAN;
#pragma unroll
    for (int i = 0; i < SPAN / 32; ++i) {
        int j = lane + 32 * i;                   // coalesced across the wave
        atomicAdd(dst + j, src[j] * s);
    }
}

// ---------------- out = relu(agg + bias) in place ----------------
__global__ void k_finalize(float* __restrict__ agg, const float* __restrict__ bias,
                           size_t total4) {
    size_t i = (size_t)blockIdx.x * blockDim.x + threadIdx.x;
    if (i >= total4) return;
    size_t idx = i * 4;
    int    f   = (int)(idx & 63);
    float4 v = *(const float4*)(agg + idx);
    v.x = fmaxf(v.x + bias[f + 0], 0.0f);
    v.y = fmaxf(v.y + bias[f + 1], 0.0f);
    v.z = fmaxf(v.z + bias[f + 2], 0.0f);
    v.w = fmaxf(v.w + bias[f + 3], 0.0f);
    *(float4*)(agg + idx) = v;
}

extern "C" void kernel_launch(void* const* d_in, const int* in_sizes, int n_in,
                              void* d_out, int out_size, void* d_ws, size_t ws_size,
                              hipStream_t stream) {
    const float*     x  = (const float*)d_in[0];      // [N, T, 64] fp32
    const long long* ei = (const long long*)d_in[1];  // [2, E] int64
    const float*     W1 = (const float*)d_in[2];
    const float*     b1 = (const float*)d_in[3];
    const float*     W2 = (const float*)d_in[4];
    const float*     b2 = (const float*)d_in[5];
    float*           out = (float*)d_out;

    const int N = NODES;
    const int E = in_sizes[1] / 2;
    const int M = N * TSTEPS;                  // 400000 rows of [*,64]

    const long long* rows = ei;                // sources
    const long long* cols = ei + E;            // targets

    // workspace: dinv[N] | h[M*64] | agg[M*64]
    float* dinv = (float*)d_ws;
    float* h    = dinv + (((size_t)N + 3) & ~(size_t)3);
    float* agg  = h + (size_t)M * 64;

    const size_t total4    = (size_t)M * 64 / 4;
    const int    vecBlocks = (int)((total4 + 255) / 256);
    const int    gemmBlk   = (M / 16 + 3) / 4;
    const int    edgeBlk   = (E + 7) / 8;

    // symmetric normalization with self-loops
    k_init_deg <<<(N + 255) / 256, 256, 0, stream>>>(dinv, N);
    k_accum_deg<<<(E + 255) / 256, 256, 0, stream>>>(cols, dinv, E);
    k_rsqrt    <<<(N + 255) / 256, 256, 0, stream>>>(dinv, N);

    // layer 1
    k_gemm64  <<<gemmBlk,  128, 0, stream>>>(x, W1, h, M);
    k_selfloop<<<vecBlocks, 256, 0, stream>>>(h, dinv, agg, total4);
    k_edge    <<<edgeBlk,  256, 0, stream>>>(rows, cols, dinv, h, agg, E);
    k_finalize<<<vecBlocks, 256, 0, stream>>>(agg, b1, total4);

    // layer 2 (accumulate straight into d_out)
    k_gemm64  <<<gemmBlk,  128, 0, stream>>>(agg, W2, h, M);
    k_selfloop<<<vecBlocks, 256, 0, stream>>>(h, dinv, out, total4);
    k_edge    <<<edgeBlk,  256, 0, stream>>>(rows, cols, dinv, h, out, E);
    k_finalize<<<vecBlocks, 256, 0, stream>>>(out, b2, total4);
}